// MDGRec_48344151883811
// MI455X (gfx1250) — compile-verified
//
#include <hip/hip_runtime.h>
#include <math.h>

// ---------------------------------------------------------------------------
// MDGRec on MI455X (gfx1250, wave32).
//   H layout: [N_NODES x 128] f32, cols 0..63 = id features, 64..127 = text.
//   ws: SUM | Ha | Hb  (each N_NODES*128 floats = 76.8 MB, total 230.4 MB)
// SpMM working set (Hcur+Hnext = 153.6 MB) fits the 192 MB L2; the read-once
// edge stream (57.6 MB/pass) and text_feats (230 MB) use NT loads so they
// don't evict it.
// ---------------------------------------------------------------------------

#define NUM_USERS 100000
#define NUM_ITEMS 50000
#define N_NODES   150000
#define EMB       64
#define TDIM      384
#define HSTRIDE   128

typedef float v2f __attribute__((ext_vector_type(2)));
typedef float v4f __attribute__((ext_vector_type(4)));
typedef float v8f __attribute__((ext_vector_type(8)));

// Non-returning CDNA5 global fp32 atomic add (tracked by STOREcnt; s_endpgm
// performs an implicit wait-idle, so no explicit s_wait_storecnt needed).
#define GATOMIC_F32(p, v, OFFSTR)                                      \
  asm volatile("global_atomic_add_f32 %0, %1, off offset:" OFFSTR      \
               :: "v"(p), "v"(v) : "memory")

// ---------------------------------------------------------------------------
// emb_id init: Hcur[:,0:64] = SUM[:,0:64] = concat(user_emb, item_emb)
// ---------------------------------------------------------------------------
__global__ void __launch_bounds__(256) init_id_kernel(
    const float* __restrict__ uemb, const float* __restrict__ iemb,
    float* __restrict__ Hcur, float* __restrict__ SUM)
{
  int idx = blockIdx.x * blockDim.x + threadIdx.x;   // v4f index
  if (idx >= N_NODES * (EMB / 4)) return;
  int node = idx >> 4;
  int c4   = idx & 15;
  v4f v = (node < NUM_USERS)
      ? ((const v4f*)uemb)[(size_t)node * 16 + c4]
      : ((const v4f*)iemb)[(size_t)(node - NUM_USERS) * 16 + c4];
  size_t o = (size_t)node * (HSTRIDE / 4) + c4;
  ((v4f*)Hcur)[o] = v;
  ((v4f*)SUM)[o]  = v;
}

// ---------------------------------------------------------------------------
// Stage a [KROWS x 64] chunk of W into LDS with K-pair interleaving:
//   sW[(k>>1)*128 + n*2 + (k&1)] = W[k][n]
// so a B fragment {W[k][n], W[k+1][n]} is one contiguous 8-byte ds_load_b64.
// ---------------------------------------------------------------------------
__device__ __forceinline__ void stage_w_pairs(float* sW, const float* Wsrc,
                                              int krows, int tid, int nthreads)
{
  int items = (krows >> 1) * (EMB / 4);       // one (kpair, n4) per item
  for (int i = tid; i < items; i += nthreads) {
    int kpair = i >> 4;                       // EMB/4 == 16
    int n = (i & 15) << 2;
    const float* r0 = Wsrc + (size_t)(kpair << 1) * EMB + n;
    v4f a = *(const v4f*)r0;
    v4f c = *(const v4f*)(r0 + EMB);
    v4f lo = {a.x, c.x, a.y, c.y};
    v4f hi = {a.z, c.z, a.w, c.w};
    float* d = sW + kpair * (2 * EMB) + (n << 1);
    *(v4f*)d       = lo;
    *(v4f*)(d + 4) = hi;
  }
}

// ---------------------------------------------------------------------------
// emb_text = text_feats @ W_text + b_text  -> Hcur[:,64:128] and SUM[:,64:128]
// 3 waves / block, one 16x64 output tile per wave via v_wmma_f32_16x16x4_f32.
// ---------------------------------------------------------------------------
#define KCHUNK 128

__global__ void __launch_bounds__(96) text_gemm_kernel(
    const float* __restrict__ text, const float* __restrict__ Wt,
    const float* __restrict__ bt, float* __restrict__ Hcur,
    float* __restrict__ SUM)
{
  __shared__ float sW[KCHUNK * EMB];   // 32 KB, pair-interleaved

  const int lane = threadIdx.x & 31;
  const int tile = blockIdx.x * 3 + (threadIdx.x >> 5);   // exact: 3125*3=9375
  const int row0 = tile * 16;

  const int m  = lane & 15;            // A row within tile / B col within tile
  const int kp = (lane >> 4) << 1;     // K pair select: 0 or 2
  const float* arow = text + (size_t)(row0 + m) * TDIM + kp;

  v8f acc0 = {}, acc1 = {}, acc2 = {}, acc3 = {};

  for (int kc = 0; kc < TDIM; kc += KCHUNK) {
    __syncthreads();
    stage_w_pairs(sW, Wt + (size_t)kc * EMB, KCHUNK, threadIdx.x, blockDim.x);
    __syncthreads();

    for (int k0 = 0; k0 < KCHUNK; k0 += 4) {
      v2f a = __builtin_nontemporal_load((const v2f*)(arow + kc + k0)); // read-once stream
      const float* bbase = sW + ((k0 + kp) >> 1) * (2 * EMB) + (m << 1);
      v2f b0 = *(const v2f*)(bbase);
      v2f b1 = *(const v2f*)(bbase + 32);
      v2f b2 = *(const v2f*)(bbase + 64);
      v2f b3 = *(const v2f*)(bbase + 96);
      acc0 = __builtin_amdgcn_wmma_f32_16x16x4_f32(false, a, false, b0, (short)0, acc0, false, false);
      acc1 = __builtin_amdgcn_wmma_f32_16x16x4_f32(false, a, false, b1, (short)0, acc1, false, false);
      acc2 = __builtin_amdgcn_wmma_f32_16x16x4_f32(false, a, false, b2, (short)0, acc2, false, false);
      acc3 = __builtin_amdgcn_wmma_f32_16x16x4_f32(false, a, false, b3, (short)0, acc3, false, false);
    }
  }

  // C/D layout: VGPR i -> M = i + 8*(lane>>4), N = (lane&15) + 16*t
  const int half = lane >> 4;
  const int n = lane & 15;
  v8f accs[4] = {acc0, acc1, acc2, acc3};
#pragma unroll
  for (int t = 0; t < 4; ++t) {
#pragma unroll
    for (int i = 0; i < 8; ++i) {
      int mm = row0 + i + 8 * half;
      int nc = t * 16 + n;
      float outv = accs[t][i] + bt[nc];
      size_t o = (size_t)mm * HSTRIDE + EMB + nc;
      Hcur[o] = outv;   // re-read by SpMM soon: keep cacheable
      SUM[o]  = outv;
    }
  }
}

// ---------------------------------------------------------------------------
// SpMM: Hout[row,:] += val * Hin[col,:]   (one wave per edge, 4 floats/lane).
// Edge stream is read-once -> NT loads; H gathers/atomics stay L2-resident.
// ---------------------------------------------------------------------------
__global__ void __launch_bounds__(256) spmm_kernel(
    const int* __restrict__ erow, const int* __restrict__ ecol,
    const float* __restrict__ ev, const float* __restrict__ Hin,
    float* __restrict__ Hout, int nEdges)
{
  int e = (blockIdx.x * blockDim.x + threadIdx.x) >> 5;
  if (e >= nEdges) return;
  int lane = threadIdx.x & 31;
  int   r = __builtin_nontemporal_load(erow + e);
  int   c = __builtin_nontemporal_load(ecol + e);
  float v = __builtin_nontemporal_load(ev + e);
  v4f h = *(const v4f*)(Hin + (size_t)c * HSTRIDE + lane * 4);
  float* dst = Hout + (size_t)r * HSTRIDE + lane * 4;
  GATOMIC_F32(dst, v * h.x, "0");
  GATOMIC_F32(dst, v * h.y, "4");
  GATOMIC_F32(dst, v * h.z, "8");
  GATOMIC_F32(dst, v * h.w, "12");
}

// ---------------------------------------------------------------------------
// SUM += H (elementwise, v4f strided)
// ---------------------------------------------------------------------------
__global__ void __launch_bounds__(256) accum_kernel(
    float* __restrict__ SUM, const float* __restrict__ H, int n4)
{
  int idx = blockIdx.x * blockDim.x + threadIdx.x;
  if (idx >= n4) return;
  v4f a = ((const v4f*)SUM)[idx];
  v4f b = ((const v4f*)H)[idx];
  ((v4f*)SUM)[idx] = a + b;
}

// ---------------------------------------------------------------------------
// Fuse: gate = sigmoid([id_final, text_final] @ W_fuse + b_fuse)
//       out  = gate*id_final + (1-gate)*text_final
// The /3 layer mean and tail amplification are folded into the A fragments
// (concat column k == SUM column k). WMMA 16x16x4 f32, K = 128.
// ---------------------------------------------------------------------------
__global__ void __launch_bounds__(96) fuse_kernel(
    const float* __restrict__ SUM, const float* __restrict__ Wf,
    const float* __restrict__ bf, const unsigned char* __restrict__ tail,
    const float* __restrict__ tamp, float* __restrict__ out)
{
  __shared__ float sW[2 * EMB * EMB];   // 128*64 floats = 32 KB, pair-interleaved
  stage_w_pairs(sW, Wf, 2 * EMB, threadIdx.x, blockDim.x);
  __syncthreads();

  const float amp  = 1.0f + 1.0f / (1.0f + __expf(-tamp[0]));
  const float inv3 = 1.0f / 3.0f;

  const int lane = threadIdx.x & 31;
  const int tile = blockIdx.x * 3 + (threadIdx.x >> 5);
  const int row0 = tile * 16;

  const int m  = lane & 15;
  const int kp = (lane >> 4) << 1;
  const int arow = row0 + m;
  const float aamp = tail[arow] ? amp : 1.0f;
  const float* srow = SUM + (size_t)arow * HSTRIDE;

  v8f acc0 = {}, acc1 = {}, acc2 = {}, acc3 = {};
  for (int k0 = 0; k0 < 2 * EMB; k0 += 4) {
    float s = (k0 < EMB) ? inv3 : inv3 * aamp;   // uniform per 4-chunk
    v2f a = *(const v2f*)(srow + k0 + kp);
    a.x *= s; a.y *= s;
    const float* bbase = sW + ((k0 + kp) >> 1) * (2 * EMB) + (m << 1);
    v2f b0 = *(const v2f*)(bbase);
    v2f b1 = *(const v2f*)(bbase + 32);
    v2f b2 = *(const v2f*)(bbase + 64);
    v2f b3 = *(const v2f*)(bbase + 96);
    acc0 = __builtin_amdgcn_wmma_f32_16x16x4_f32(false, a, false, b0, (short)0, acc0, false, false);
    acc1 = __builtin_amdgcn_wmma_f32_16x16x4_f32(false, a, false, b1, (short)0, acc1, false, false);
    acc2 = __builtin_amdgcn_wmma_f32_16x16x4_f32(false, a, false, b2, (short)0, acc2, false, false);
    acc3 = __builtin_amdgcn_wmma_f32_16x16x4_f32(false, a, false, b3, (short)0, acc3, false, false);
  }

  const int half = lane >> 4;
  const int n = lane & 15;
  v8f accs[4] = {acc0, acc1, acc2, acc3};
#pragma unroll
  for (int t = 0; t < 4; ++t) {
#pragma unroll
    for (int i = 0; i < 8; ++i) {
      int mm = row0 + i + 8 * half;
      int nc = t * 16 + n;
      float g = 1.0f / (1.0f + __expf(-(accs[t][i] + bf[nc])));
      float ta  = tail[mm] ? amp : 1.0f;
      float idv = SUM[(size_t)mm * HSTRIDE + nc] * inv3;
      float txv = SUM[(size_t)mm * HSTRIDE + EMB + nc] * inv3 * ta;
      __builtin_nontemporal_store(g * idv + (1.0f - g) * txv,
                                  out + (size_t)mm * EMB + nc);
    }
  }
}

// ---------------------------------------------------------------------------
extern "C" void kernel_launch(void* const* d_in, const int* in_sizes, int n_in,
                              void* d_out, int out_size, void* d_ws, size_t ws_size,
                              hipStream_t stream)
{
  const float*         text = (const float*)d_in[0];
  const int*           erow = (const int*)d_in[1];
  const int*           ecol = (const int*)d_in[2];
  const float*         ev   = (const float*)d_in[3];
  const unsigned char* tail = (const unsigned char*)d_in[4]; // jnp bool = 1 byte
  const float*         uemb = (const float*)d_in[5];
  const float*         iemb = (const float*)d_in[6];
  const float*         Wt   = (const float*)d_in[7];
  const float*         bt   = (const float*)d_in[8];
  const float*         Wf   = (const float*)d_in[9];
  const float*         bf   = (const float*)d_in[10];
  const float*         tamp = (const float*)d_in[11];
  float* out = (float*)d_out;
  const int nEdges = in_sizes[1];

  const size_t perBuf = (size_t)N_NODES * HSTRIDE;   // floats
  float* SUM = (float*)d_ws;
  float* Ha  = SUM + perBuf;
  float* Hb  = Ha + perBuf;

  // Layer-0 embeddings into Ha (=h0) and SUM (id half + text half).
  {
    int n4 = N_NODES * (EMB / 4);
    init_id_kernel<<<(n4 + 255) / 256, 256, 0, stream>>>(uemb, iemb, Ha, SUM);
  }
  {
    int blocks = (N_NODES / 16) / 3;   // 9375 tiles / 3 waves = 3125, exact
    text_gemm_kernel<<<blocks, 96, 0, stream>>>(text, Wt, bt, Ha, SUM);
  }

  const long long spmmThreads = (long long)nEdges * 32;
  const int spmmBlocks = (int)((spmmThreads + 255) / 256);

  // Layer 1: Hb = A @ Ha ; SUM += Hb.
  hipMemsetAsync(Hb, 0, perBuf * sizeof(float), stream);
  spmm_kernel<<<spmmBlocks, 256, 0, stream>>>(erow, ecol, ev, Ha, Hb, nEdges);
  {
    int n4 = (int)(perBuf / 4);
    accum_kernel<<<(n4 + 255) / 256, 256, 0, stream>>>(SUM, Hb, n4);
  }

  // Layer 2: scatter-add A @ Hb directly into SUM (skips memset + accum).
  spmm_kernel<<<spmmBlocks, 256, 0, stream>>>(erow, ecol, ev, Hb, SUM, nEdges);

  // Gated fusion (WMMA GEMM + sigmoid) -> out.
  {
    int blocks = (N_NODES / 16) / 3;
    fuse_kernel<<<blocks, 96, 0, stream>>>(SUM, Wf, bf, tail, tamp, out);
  }
}